// Ranking_model_43696997269789
// MI455X (gfx1250) — compile-verified
//
#include <hip/hip_runtime.h>
#include <math.h>

#define B_    64
#define N_    2048
#define EMB_  128
#define LCONV 126   // EMB - 3 + 1
#define K2_   64
#define WOUT  63

typedef float v8f __attribute__((ext_vector_type(8)));
typedef float v2f __attribute__((ext_vector_type(2)));

// ---------------------------------------------------------------------------
// Prep kernel: collapse FC stack to (v, c0); per-b r1 conv, t1[w] (r1's c3
// contribution + b3), cnt[b] (first zero uid), high scores; zero d_out.
// ---------------------------------------------------------------------------
__global__ __launch_bounds__(256) void prep_kernel(
    const float* __restrict__ top_user, const float* __restrict__ question,
    const int*   __restrict__ uids,
    const float* __restrict__ W1, const float* __restrict__ b1,
    const float* __restrict__ W2, const float* __restrict__ b2,
    const float* __restrict__ W3, const float* __restrict__ b3,
    const float* __restrict__ fw1, const float* __restrict__ fb1,
    const float* __restrict__ fw2, const float* __restrict__ fb2,
    const float* __restrict__ fw3, const float* __restrict__ fb3,
    float* __restrict__ ws_v, float* __restrict__ ws_c0,
    float* __restrict__ ws_hs, int* __restrict__ ws_cnt,
    float* __restrict__ ws_r1, float* __restrict__ ws_t1,
    float* __restrict__ out)
{
  __shared__ float gsh[16];
  __shared__ float vsh[64];
  __shared__ float c0sh;
  const int t = threadIdx.x;

  if (t < 16) { float a = 0.f; for (int j = 0; j < 16; ++j) a += fw3[j] * fw2[j*16 + t]; gsh[t] = a; }
  __syncthreads();

  if (t < 64) {
    float val = 0.f;
    if (t < WOUT) { for (int m = 0; m < 16; ++m) val += gsh[m] * fw1[m*WOUT + t]; }
    vsh[t] = val; ws_v[t] = val;           // v[63] = 0 pad
  }
  if (t == 0) {
    float c0 = fb3[0];
    for (int m = 0; m < 16; ++m) c0 += gsh[m] * fb1[m];
    for (int j = 0; j < 16; ++j) c0 += fw3[j] * fb2[j];
    c0sh = c0; ws_c0[0] = c0;
    out[0] = 0.f;
  }
  if (t < 64) {                             // cnt[b]: first zero uid else N
    int c = N_;
    const int* row = uids + t * N_;
    for (int n = 0; n < N_; ++n) { if (row[n] == 0) { c = n; break; } }
    ws_cnt[t] = c;
  }
  {                                         // r1[b][x] = relu(conv3(q))
    const float w10 = W1[0], w11 = W1[1], w12 = W1[2], b1v = b1[0];
    const int b = t >> 2, part = t & 3;
    const float* q = question + b * EMB_;
    for (int k = 0; k < 32; ++k) {
      int x = part * 32 + k;
      float r = 0.f;
      if (x < LCONV) r = fmaxf(fmaf(q[x], w10, fmaf(q[x+1], w11, fmaf(q[x+2], w12, b1v))), 0.f);
      ws_r1[b * EMB_ + x] = r;
    }
  }
  __syncthreads();
  {                                         // t1[b][w] = b3 + sum_j r1[w+j]*W3[1,j]
    const float b3v = b3[0];
    const int b = t >> 2, part = t & 3;
    const float* r1p = ws_r1 + b * EMB_;
    for (int k = 0; k < 16; ++k) {
      int w = part * 16 + k;
      float a = b3v;
      if (w < WOUT) { for (int j = 0; j < K2_; ++j) a += r1p[w + j] * W3[64 + j]; }
      ws_t1[b * 64 + w] = a;
    }
  }
  if (t < 64) {                             // high score per b (scalar path)
    const float w10 = W1[0], w11 = W1[1], w12 = W1[2], b1v = b1[0];
    const float w20 = W2[0], w21 = W2[1], w22 = W2[2];
    const float w23 = W2[3], w24 = W2[4], w25 = W2[5], b2v = b2[0];
    const float b3v = b3[0];
    const float* u = top_user + t * EMB_;
    const float* q = question + t * EMB_;
    float r0a[LCONV], r1a[LCONV], r2a[LCONV];
    for (int x = 0; x < LCONV; ++x) {
      r0a[x] = fmaxf(fmaf(u[x], w10, fmaf(u[x+1], w11, fmaf(u[x+2], w12, b1v))), 0.f);
      r1a[x] = fmaxf(fmaf(q[x], w10, fmaf(q[x+1], w11, fmaf(q[x+2], w12, b1v))), 0.f);
      r2a[x] = fmaxf(fmaf(u[x], w20, fmaf(u[x+1], w21, fmaf(u[x+2], w22,
               fmaf(q[x], w23, fmaf(q[x+1], w24, fmaf(q[x+2], w25, b2v)))))), 0.f);
    }
    float s = c0sh;
    for (int w = 0; w < WOUT; ++w) {
      float a = b3v;
      for (int j = 0; j < K2_; ++j)
        a += r0a[w+j] * W3[j] + r1a[w+j] * W3[64 + j] + r2a[w+j] * W3[128 + j];
      s += fmaxf(a, 0.f) * vsh[w];
    }
    ws_hs[t] = s;
  }
}

// ---------------------------------------------------------------------------
// Main kernel: 64 threads = 2 waves; each wave scores 16 (b,n) pairs via
// banded-matmul on V_WMMA_F32_16X16X4_F32 (160 WMMA per wave).
// ---------------------------------------------------------------------------
__global__ __launch_bounds__(64) void score_kernel(
    const float* __restrict__ users, const float* __restrict__ question,
    const float* __restrict__ W1, const float* __restrict__ b1,
    const float* __restrict__ W2, const float* __restrict__ b2,
    const float* __restrict__ W3, const float* __restrict__ b3,
    const float* __restrict__ ws_v, const float* __restrict__ ws_c0,
    const float* __restrict__ ws_hs, const int* __restrict__ ws_cnt,
    const float* __restrict__ ws_t1, float* __restrict__ out)
{
  constexpr int ROWSTR  = 134;   // floats per r-row (128 used + pad)
  constexpr int PAIRSTR = 268;   // 2 rows per pair; 268 % 64 banks -> conflict-free

  __shared__ __align__(16) float rbuf[2][16 * PAIRSTR]; // [wave][pair: r0,r2]
  __shared__ __align__(16) v2f   Bw[2][20][32];         // banded W3 in B-operand layout
  __shared__ __align__(16) float qs[132];
  __shared__ __align__(16) float t1s[64];
  __shared__ __align__(16) float vsh[64];

  const int b    = blockIdx.y;
  const int n0   = blockIdx.x * 32;
  const int t    = threadIdx.x;
  const int wave = t >> 5, lane = t & 31;

  for (int k = t; k < 128; k += 64) qs[k] = question[b * EMB_ + k];
  if (t < 4)  qs[128 + t] = 0.f;
  if (t < 64) { vsh[t] = ws_v[t]; t1s[t] = ws_t1[b * 64 + t]; }

  // B[k,n] = W3[row][4*ks + koff + c - n] (0 if outside [0,64)); identical for
  // all four 16-wide output tiles, so built once. Operand layout: lane L holds
  // column n = L%16, K-pair koff = 2*(L>=16).
  for (int e = t; e < 1280; e += 64) {
    int ln = e & 31; int tmp = e >> 5; int ks = tmp % 20; int i = tmp / 20; // i: 0->r0, 1->r2
    int nn = ln & 15; int k0 = (ln >> 4) * 2;
    int wrow = (i == 0) ? 0 : 2;
    v2f val;
    #pragma unroll
    for (int c = 0; c < 2; ++c) {
      int j = 4 * ks + k0 + c - nn;
      val[c] = (j >= 0 && j < 64) ? W3[wrow * 64 + j] : 0.f;
    }
    Bw[i][ks][ln] = val;
  }

  const float w10 = W1[0], w11 = W1[1], w12 = W1[2], b1v = b1[0];
  const float w20 = W2[0], w21 = W2[1], w22 = W2[2];
  const float w23 = W2[3], w24 = W2[4], w25 = W2[5], b2v = b2[0];
  const float c0v = ws_c0[0];
  const float hs  = ws_hs[b];
  const int   cntb = ws_cnt[b];
  __syncthreads();

  // ---- r0 / r2 for this wave's 16 pairs (coalesced global streaming) ----
  const float* ubase = users + ((size_t)b * N_ + n0 + wave * 16) * EMB_;
  for (int it = 0; it < 8; ++it) {
    int e = lane + 32 * it;                 // 16 pairs x 16 blocks of 8 x-values
    int pair = e >> 4, xb = e & 15, x0 = xb * 8;
    const float* up = ubase + pair * EMB_ + x0;
    __builtin_prefetch(up + 2 * EMB_, 0, 0);
    float uv[10];
    float4 a4 = *(const float4*)(up);
    float4 b4 = *(const float4*)(up + 4);
    uv[0] = a4.x; uv[1] = a4.y; uv[2] = a4.z; uv[3] = a4.w;
    uv[4] = b4.x; uv[5] = b4.y; uv[6] = b4.z; uv[7] = b4.w;
    if (x0 + 9 < EMB_) { float2 c2 = *(const float2*)(up + 8); uv[8] = c2.x; uv[9] = c2.y; }
    else               { uv[8] = 0.f; uv[9] = 0.f; }
    float* dst = &rbuf[wave][pair * PAIRSTR];
    #pragma unroll
    for (int c = 0; c < 8; ++c) {
      int x = x0 + c;
      float r0v = 0.f, r2v = 0.f;
      if (x < LCONV) {
        r0v = fmaxf(fmaf(uv[c], w10, fmaf(uv[c+1], w11, fmaf(uv[c+2], w12, b1v))), 0.f);
        r2v = fmaxf(fmaf(uv[c], w20, fmaf(uv[c+1], w21, fmaf(uv[c+2], w22,
              fmaf(qs[x], w23, fmaf(qs[x+1], w24, fmaf(qs[x+2], w25, b2v)))))), 0.f);
      }
      dst[x]          = r0v;
      dst[ROWSTR + x] = r2v;
    }
  }
  __syncthreads();

  // ---- banded matmul on WMMA f32 16x16x4: M=pairs, N=w-tile, K=x-window ----
  v8f acc0 = {}, acc1 = {}, acc2 = {}, acc3 = {};
  const int koff = (lane >> 4) * 2;         // A layout: lane holds M=lane%16, K=koff..koff+1
  const float* a0p = &rbuf[wave][(lane & 15) * PAIRSTR] + koff;
  for (int i = 0; i < 2; ++i) {
    const float* rp = a0p + i * ROWSTR;     // i=0: r0 row, i=1: r2 row
    for (int ks = 0; ks < 20; ++ks) {
      v2f bv = Bw[i][ks][lane];
      const float* pb = rp + 4 * ks;
      v2f av0 = *(const v2f*)(pb);
      v2f av1 = *(const v2f*)(pb + 16);
      v2f av2 = *(const v2f*)(pb + 32);
      v2f av3 = *(const v2f*)(pb + 48);
      acc0 = __builtin_amdgcn_wmma_f32_16x16x4_f32(false, av0, false, bv, (short)0, acc0, false, false);
      acc1 = __builtin_amdgcn_wmma_f32_16x16x4_f32(false, av1, false, bv, (short)0, acc1, false, false);
      acc2 = __builtin_amdgcn_wmma_f32_16x16x4_f32(false, av2, false, bv, (short)0, acc2, false, false);
      acc3 = __builtin_amdgcn_wmma_f32_16x16x4_f32(false, av3, false, bv, (short)0, acc3, false, false);
    }
  }

  // ---- epilogue: +t1(+b3), relu, dot v, sigmoid, mask, reduce ----
  float total = 0.f;
  const int hi = lane >> 4, nlo = lane & 15;
  const float t0w = t1s[nlo],      t1w = t1s[16 + nlo];
  const float t2w = t1s[32 + nlo], t3w = t1s[48 + nlo];
  const float v0w = vsh[nlo],      v1w = vsh[16 + nlo];
  const float v2w = vsh[32 + nlo], v3w = vsh[48 + nlo]; // vsh[63]=0 kills pad col
  #pragma unroll
  for (int r = 0; r < 8; ++r) {
    float p = fmaxf(acc0[r] + t0w, 0.f) * v0w
            + fmaxf(acc1[r] + t1w, 0.f) * v1w
            + fmaxf(acc2[r] + t2w, 0.f) * v2w
            + fmaxf(acc3[r] + t3w, 0.f) * v3w;
    p += __shfl_xor(p, 1, 32);
    p += __shfl_xor(p, 2, 32);
    p += __shfl_xor(p, 4, 32);
    p += __shfl_xor(p, 8, 32);              // reduced within each 16-lane half
    if (nlo == r) {
      int m = r + 8 * hi;                   // C layout: VGPR r -> pair r / r+8
      int n = n0 + wave * 16 + m;
      if (n < cntb) {
        float z = (p + c0v) - hs;
        total += 1.f / (1.f + expf(-z));
      }
    }
  }
  total += __shfl_xor(total, 1, 32);
  total += __shfl_xor(total, 2, 32);
  total += __shfl_xor(total, 4, 32);
  total += __shfl_xor(total, 8, 32);
  total += __shfl_xor(total, 16, 32);
  if (lane == 0) atomicAdd(out, total);
}

// ---------------------------------------------------------------------------
extern "C" void kernel_launch(void* const* d_in, const int* in_sizes, int n_in,
                              void* d_out, int out_size, void* d_ws, size_t ws_size,
                              hipStream_t stream) {
  const float* users    = (const float*)d_in[0];
  const float* top_user = (const float*)d_in[1];
  const float* question = (const float*)d_in[2];
  const int*   uids     = (const int*)  d_in[3];
  const float* W1  = (const float*)d_in[4];
  const float* b1  = (const float*)d_in[5];
  const float* W2  = (const float*)d_in[6];
  const float* b2  = (const float*)d_in[7];
  const float* W3  = (const float*)d_in[8];
  const float* b3  = (const float*)d_in[9];
  const float* fw1 = (const float*)d_in[10];
  const float* fb1 = (const float*)d_in[11];
  const float* fw2 = (const float*)d_in[12];
  const float* fb2 = (const float*)d_in[13];
  const float* fw3 = (const float*)d_in[14];
  const float* fb3 = (const float*)d_in[15];
  float* out = (float*)d_out;

  float* ws     = (float*)d_ws;
  float* ws_v   = ws;                 // 64
  float* ws_c0  = ws + 64;            // 1
  float* ws_hs  = ws + 80;            // 64
  int*   ws_cnt = (int*)(ws + 160);   // 64
  float* ws_r1  = ws + 256;           // 64*128
  float* ws_t1  = ws + 256 + 64*128;  // 64*64

  prep_kernel<<<1, 256, 0, stream>>>(top_user, question, uids,
                                     W1, b1, W2, b2, W3, b3,
                                     fw1, fb1, fw2, fb2, fw3, fb3,
                                     ws_v, ws_c0, ws_hs, ws_cnt, ws_r1, ws_t1, out);
  score_kernel<<<dim3(N_ / 32, B_), 64, 0, stream>>>(users, question,
                                     W1, b1, W2, b2, W3, b3,
                                     ws_v, ws_c0, ws_hs, ws_cnt, ws_t1, out);
}